// GCNLayer_9689446220544
// MI455X (gfx1250) — compile-verified
//
#include <hip/hip_runtime.h>

typedef __attribute__((ext_vector_type(2))) float v2f;
typedef __attribute__((ext_vector_type(8))) float v8f;

#define N_NODES 50000
#define N_EDGES 600000
#define D 128

// ---------------- zero workspace (h1 + h2) ----------------
__global__ void gcn_zero_kernel(float* __restrict__ p, long n) {
    long i = (long)blockIdx.x * blockDim.x + threadIdx.x;
    long stride = (long)gridDim.x * blockDim.x;
    for (; i < n; i += stride) p[i] = 0.0f;
}

// ---------------- edge scatter-add: y[dst[e], :] += x[src[e], :] ----------------
// thread <-> (edge, column); a wave covers 32 consecutive columns of one edge,
// so both the gather read and the atomic destination are coalesced 128B bursts.
__global__ void gcn_scatter_add_kernel(const float* __restrict__ x,
                                       const int* __restrict__ src,
                                       const int* __restrict__ dst,
                                       float* __restrict__ y) {
    long i = (long)blockIdx.x * blockDim.x + threadIdx.x;
    const long total = (long)N_EDGES * D;
    if (i >= total) return;
    int e = (int)(i >> 7);      // / 128
    int c = (int)(i & (D - 1)); // % 128
    int s = src[e];
    int d = dst[e];
    atomicAdd(&y[(long)d * D + c], x[(long)s * D + c]);
}

// ---------------- out = A @ W + b via V_WMMA_F32_16X16X4_F32 ----------------
// One wave per 16x16 output tile. Block = 8 waves = the 8 N-tiles of one
// 16-row stripe. Grid = 50000/16 = 3125 stripes (exact, no edge handling).
__global__ void __launch_bounds__(256)
gcn_gemm_bias_kernel(const float* __restrict__ A,   // [50000,128] (h2)
                     const float* __restrict__ W,   // [128,128]
                     const float* __restrict__ bias,// [128]
                     float* __restrict__ out) {     // [50000,128]
    const int mTile = blockIdx.x;
    const int wave  = threadIdx.x >> 5;   // 0..7 -> N tile
    const int lane  = threadIdx.x & 31;
    const int row16 = lane & 15;          // M (A), N (B/C/D) sub-index
    const int khalf = lane >> 4;          // selects K pair / output row half
    const int mBase = mTile * 16;
    const int nBase = wave * 16;

    const float* aRow = A + (long)(mBase + row16) * D;  // this lane's A row
    const float* wCol = W + nBase + row16;              // this lane's B column

    v8f c = {};
#pragma unroll 8
    for (int k0 = 0; k0 < D; k0 += 4) {
        const int ka = k0 + khalf * 2;
        // A 16x4 fp32 layout: lanes 0-15 -> K = k0,k0+1 ; lanes 16-31 -> K = k0+2,k0+3
        v2f a;
        a.x = aRow[ka];
        a.y = aRow[ka + 1];
        // B 4x16 fp32 layout: lane%16 = N ; lane/16 selects K half
        v2f bm;
        bm.x = wCol[(long)ka * D];
        bm.y = wCol[(long)(ka + 1) * D];
        // 8 args: (neg_a, A, neg_b, B, c_mod, C, reuse_a, reuse_b)
        c = __builtin_amdgcn_wmma_f32_16x16x4_f32(false, a, false, bm,
                                                  (short)0, c, false, false);
    }

    const float bv = bias[nBase + row16];
#pragma unroll
    for (int i = 0; i < 8; ++i) {
        // C/D layout: VGPR i = row (i + 8*khalf), col = lane%16
        const int m = mBase + khalf * 8 + i;
        out[(long)m * D + nBase + row16] = c[i] + bv;
    }
}

extern "C" void kernel_launch(void* const* d_in, const int* in_sizes, int n_in,
                              void* d_out, int out_size, void* d_ws, size_t ws_size,
                              hipStream_t stream) {
    const float* feature = (const float*)d_in[0];
    const float* W       = (const float*)d_in[1];
    const float* bias    = (const float*)d_in[2];
    const int*   src     = (const int*)d_in[3];
    const int*   dst     = (const int*)d_in[4];
    float* out = (float*)d_out;

    float* h1 = (float*)d_ws;                    // [50000,128]
    float* h2 = h1 + (long)N_NODES * D;          // [50000,128]

    // zero h1 and h2 (contiguous) every call — deterministic
    const long zn = 2L * N_NODES * D;
    gcn_zero_kernel<<<2048, 256, 0, stream>>>(h1, zn);

    const long edgeWork = (long)N_EDGES * D;     // 76.8M threads
    const int  eBlocks  = (int)((edgeWork + 255) / 256);
    gcn_scatter_add_kernel<<<eBlocks, 256, 0, stream>>>(feature, src, dst, h1);
    gcn_scatter_add_kernel<<<eBlocks, 256, 0, stream>>>(h1, src, dst, h2);

    gcn_gemm_bias_kernel<<<N_NODES / 16, 256, 0, stream>>>(h2, W, bias, out);
}